// GNN_53145925321329
// MI455X (gfx1250) — compile-verified
//
#include <hip/hip_runtime.h>
#include <math.h>

typedef __attribute__((ext_vector_type(2))) float v2f;
typedef __attribute__((ext_vector_type(8))) float v8f;

// Workspace layout (floats):
//   [0..63]    u       (B=8 x D=8)
//   [64..159]  rel_w   (B=8 x R=12)
//   [160..223] acc     (B=8 x D=8) -- zeroed by prep each launch
#define WS_U    0
#define WS_RELW 64
#define WS_ACC  160

// ---------------------------------------------------------------------------
// Kernel 1: gather user embeddings, compute rel_w = u @ relation_emb^T,
// zero the accumulator. One tiny block.
// ---------------------------------------------------------------------------
__global__ void gnn_prep(const int* __restrict__ user,
                         const float* __restrict__ user_emb,
                         const float* __restrict__ rel_emb,
                         float* __restrict__ ws) {
    int t = threadIdx.x;                       // 128 threads
    if (t < 64) {
        int b = t >> 3, d = t & 7;
        ws[WS_U + t]   = user_emb[(long)user[b] * 8 + d];
        ws[WS_ACC + t] = 0.0f;                 // fresh accumulator every launch
    }
    if (t < 96) {
        int b = t / 12, r = t - b * 12;
        const float* ub = user_emb + (long)user[b] * 8;
        const float* rr = rel_emb + r * 8;
        float s = 0.0f;
#pragma unroll
        for (int d = 0; d < 8; ++d) s += ub[d] * rr[d];
        ws[WS_RELW + t] = s;                   // rel_w[b][r]
    }
}

// ---------------------------------------------------------------------------
// Kernel 2: stream rows[] (6.4 MB); for the ~16 edges hitting segment e0,
// accumulate rel_w[b, values[i]] * entity_emb[cols[i], :] into acc[b][:].
// ---------------------------------------------------------------------------
__global__ void gnn_scan(const int* __restrict__ entity,
                         const int* __restrict__ values,
                         const int* __restrict__ rows,
                         const int* __restrict__ cols,
                         const float* __restrict__ ent_emb,
                         float* __restrict__ ws,
                         int E) {
    int i = blockIdx.x * blockDim.x + threadIdx.x;
    if (i >= E) return;
    int e0 = entity[0];                        // uniform -> scalar load
    if (rows[i] != e0) return;                 // ~16/1.6M survive

    int val = values[i];
    const float* src = ent_emb + (long)cols[i] * 8;
    float s[8];
#pragma unroll
    for (int d = 0; d < 8; ++d) s[d] = src[d];

    const float* relw = ws + WS_RELW;
    float* acc = ws + WS_ACC;
#pragma unroll
    for (int b = 0; b < 8; ++b) {
        float w = relw[b * 12 + val];
#pragma unroll
        for (int d = 0; d < 8; ++d)
            __hip_atomic_fetch_add(&acc[b * 8 + d], w * s[d],
                                   __ATOMIC_RELAXED, __HIP_MEMORY_SCOPE_AGENT);
    }
}

// ---------------------------------------------------------------------------
// Kernel 3 (one wave32): l0 = acc(8x8) @ W(8x8) via two chained
// V_WMMA_F32_16X16X4_F32 (K = 0..3 then 4..7), then out = u * tanh(l0).
//
// A layout (32-bit A 16x4): lanes 0-15 hold K={0,1} in v[0],v[1];
//                           lanes 16-31 hold K={2,3}.       (ISA 7.12.2)
// B layout (rows striped across lanes in a VGPR):
//   v[0]: lanes 0-15 = row K, N=lane; lanes 16-31 = row K+2, N=lane-16.
// C/D layout: VGPR r, lanes 0-15 -> D[r][lane]  (valid region M,N < 8).
// ---------------------------------------------------------------------------
__global__ void gnn_finish(const float* __restrict__ ws,
                           const float* __restrict__ W,
                           float* __restrict__ out) {
    int lane = threadIdx.x;
    int half = lane >> 4;                      // selects K pair
    int m    = lane & 15;                      // A row / B column

    float a0 = 0.f, a1 = 0.f, a2 = 0.f, a3 = 0.f;
    float b0 = 0.f, b1 = 0.f, b2 = 0.f, b3 = 0.f;
    if (m < 8) {
        const float* ap = ws + WS_ACC + m * 8;         // acc row m
        a0 = ap[half * 2 + 0];      a1 = ap[half * 2 + 1];       // K 0..3
        a2 = ap[4 + half * 2 + 0];  a3 = ap[4 + half * 2 + 1];   // K 4..7
        b0 = W[(half * 2 + 0) * 8 + m];                          // W row K, col m
        b1 = W[(half * 2 + 1) * 8 + m];
        b2 = W[(4 + half * 2 + 0) * 8 + m];
        b3 = W[(4 + half * 2 + 1) * 8 + m];
    }

    v8f c = {};
    v2f A1 = {a0, a1}, B1 = {b0, b1};
    c = __builtin_amdgcn_wmma_f32_16x16x4_f32(false, A1, false, B1,
                                              (short)0, c, false, false);
    v2f A2 = {a2, a3}, B2 = {b2, b3};
    c = __builtin_amdgcn_wmma_f32_16x16x4_f32(false, A2, false, B2,
                                              (short)0, c, false, false);

    if (lane < 8) {                            // lane == output column d
#pragma unroll
        for (int b = 0; b < 8; ++b)
            out[b * 8 + lane] = ws[WS_U + b * 8 + lane] * tanhf(c[b]);
    }
}

// ---------------------------------------------------------------------------
extern "C" void kernel_launch(void* const* d_in, const int* in_sizes, int n_in,
                              void* d_out, int out_size, void* d_ws, size_t ws_size,
                              hipStream_t stream) {
    const int*   user     = (const int*)  d_in[0];
    const int*   entity   = (const int*)  d_in[1];
    const int*   values   = (const int*)  d_in[2];
    const int*   indices  = (const int*)  d_in[3];   // (2, E) flat
    const float* user_emb = (const float*)d_in[4];
    const float* rel_emb  = (const float*)d_in[5];
    const float* ent_emb  = (const float*)d_in[6];
    const float* W        = (const float*)d_in[7];

    int E = in_sizes[2];
    const int* rows = indices;
    const int* cols = indices + E;
    float* ws = (float*)d_ws;

    hipLaunchKernelGGL(gnn_prep, dim3(1), dim3(128), 0, stream,
                       user, user_emb, rel_emb, ws);

    int blocks = (E + 255) / 256;
    hipLaunchKernelGGL(gnn_scan, dim3(blocks), dim3(256), 0, stream,
                       entity, values, rows, cols, ent_emb, ws, E);

    hipLaunchKernelGGL(gnn_finish, dim3(1), dim3(32), 0, stream,
                       ws, W, (float*)d_out);
}